// SpatialTransformer_43147241456355
// MI455X (gfx1250) — compile-verified
//
#include <hip/hip_runtime.h>
#include <stdint.h>

// Spatial transformer (affine grid + bilinear sample), N=8 C=32 H=W=512, f32.
// Memory-bound: 512 MiB traffic -> ~22us floor at 23.3 TB/s. Coordinate math is
// shared across all N*C=256 planes, so compute it once per pixel and loop planes.
// CDNA5 path: theta is DMA'd to LDS via the Tensor Data Mover (tensor_load_to_lds
// + s_wait_tensorcnt); output uses non-temporal stores to preserve L2 for input.

#define ST_N 8
#define ST_C 32
#define ST_H 512
#define ST_W 512
#define ST_NC (ST_N * ST_C)          // 256 planes
#define PLANE_ELEMS (ST_H * ST_W)    // 262144 = 1<<18

typedef uint32_t u32x4 __attribute__((ext_vector_type(4)));
typedef int32_t  i32x8 __attribute__((ext_vector_type(8)));
typedef int32_t  i32x4 __attribute__((ext_vector_type(4)));

__global__ __launch_bounds__(256)
void SpatialTransformer_43147241456355_kernel(const float* __restrict__ x,
                                              const float* __restrict__ theta,
                                              float* __restrict__ out) {
  __shared__ float sT[16];   // theta staging (6 floats used), 64B-aligned block

  // ---- TDM: DMA the 6 theta floats into LDS (wave 0 only) -----------------
  const int flat = threadIdx.x + 64 * threadIdx.y;
  if (flat < 32) {                           // exactly wave 0 of this block
    const uint64_t ga  = (uint64_t)(uintptr_t)theta;
    const uint32_t lds = (uint32_t)(uintptr_t)(&sT[0]);   // flat low 32b = LDS offset
    // D# group0: count=1 | lds_addr | global_addr[56:0] | type=2 ("image")
    u32x4 g0 = { 1u,
                 lds,
                 (uint32_t)ga,
                 (uint32_t)((ga >> 32) & 0x1FFFFFFu) | (2u << 30) };
    // D# group1: data_size=4B(code 2) @bits17:16 ; tensor_dim0=6 @bits79:48 ;
    //            tensor_dim1=1 @bits111:80 ; tile_dim0=6 @bits127:112 ;
    //            tile_dim1=1 @bits143:128 ; tensor_dim0_stride=6 @bits207:160
    i32x8 g1 = { (int)(2u << 16),
                 (int)(6u << 16),     // dim0 low bits -> [63:48]
                 (int)(1u << 16),     // dim1 low bits -> [95:80]
                 (int)(6u << 16),     // tile_dim0 -> [127:112]
                 1,                   // tile_dim1 -> [143:128]
                 6,                   // stride0 low -> [191:160]
                 0, 0 };
    i32x4 g2 = { 0, 0, 0, 0 };
    i32x4 g3 = { 0, 0, 0, 0 };
    i32x8 g4 = { 0, 0, 0, 0, 0, 0, 0, 0 };   // 6-arg toolchain: extra group
    __builtin_amdgcn_tensor_load_to_lds(g0, g1, g2, g3, g4, /*cpol=*/0);
    __builtin_amdgcn_s_wait_tensorcnt(0);
  }
  __syncthreads();

  const float t00 = sT[0], t01 = sT[1], t02 = sT[2];
  const float t10 = sT[3], t11 = sT[4], t12 = sT[5];

  // ---- Per-pixel coordinate math (done once, reused for all planes) -------
  const int pix = blockIdx.x * 64 + threadIdx.x;   // 0 .. 262143
  const int h = pix >> 9;          // W = 512
  const int w = pix & (ST_W - 1);

  const float step = 2.0f / (float)(ST_W - 1);     // H == W
  const float xs = fmaf((float)w, step, -1.0f);
  const float ys = fmaf((float)h, step, -1.0f);
  const float gx = fmaf(t00, xs, fmaf(t01, ys, t02));
  const float gy = fmaf(t10, xs, fmaf(t11, ys, t12));
  const float ix = (gx + 1.0f) * (0.5f * (float)(ST_W - 1));
  const float iy = (gy + 1.0f) * (0.5f * (float)(ST_H - 1));

  const float x0f = floorf(ix), y0f = floorf(iy);
  const int x0 = (int)x0f, y0 = (int)y0f;
  const int x1 = x0 + 1,   y1 = y0 + 1;

  float wx1 = ix - x0f, wx0 = 1.0f - wx1;
  float wy1 = iy - y0f, wy0 = 1.0f - wy1;
  // Fold per-corner validity into the weights (mask * weight, exact 0/1 mult).
  wx0 = (x0 >= 0 && x0 < ST_W) ? wx0 : 0.0f;
  wx1 = (x1 >= 0 && x1 < ST_W) ? wx1 : 0.0f;
  wy0 = (y0 >= 0 && y0 < ST_H) ? wy0 : 0.0f;
  wy1 = (y1 >= 0 && y1 < ST_H) ? wy1 : 0.0f;
  const float w00 = wy0 * wx0, w01 = wy0 * wx1;
  const float w10 = wy1 * wx0, w11 = wy1 * wx1;

  const int xc0 = min(max(x0, 0), ST_W - 1), xc1 = min(max(x1, 0), ST_W - 1);
  const int yc0 = min(max(y0, 0), ST_H - 1), yc1 = min(max(y1, 0), ST_H - 1);
  const int o00 = yc0 * ST_W + xc0, o01 = yc0 * ST_W + xc1;
  const int o10 = yc1 * ST_W + xc0, o11 = yc1 * ST_W + xc1;

  // ---- Plane loop: 4 gathers + 4 FMAs + 1 NT store per element ------------
  // planes handled by this thread: blockIdx.y*32 + threadIdx.y + 4*i, i<8
  int planeBase = (blockIdx.y * 32 + threadIdx.y) << 18;
#pragma unroll
  for (int i = 0; i < 8; ++i) {
    const float* p = x + planeBase;
    float acc = w00 * p[o00];
    acc = fmaf(w01, p[o01], acc);
    acc = fmaf(w10, p[o10], acc);
    acc = fmaf(w11, p[o11], acc);
    __builtin_nontemporal_store(acc, out + planeBase + pix);
    planeBase += (4 << 18);
  }
}

extern "C" void kernel_launch(void* const* d_in, const int* in_sizes, int n_in,
                              void* d_out, int out_size, void* d_ws, size_t ws_size,
                              hipStream_t stream) {
  const float* x     = (const float*)d_in[0];
  const float* theta = (const float*)d_in[1];
  float* out         = (float*)d_out;

  // 4096 strips of 64 pixels; plane dimension split 8 ways (32 planes/block).
  dim3 grid(PLANE_ELEMS / 64, ST_NC / 32);
  dim3 block(64, 4);
  hipLaunchKernelGGL(SpatialTransformer_43147241456355_kernel,
                     grid, block, 0, stream, x, theta, out);
}